// TwoStepPredictor_54047868453216
// MI455X (gfx1250) — compile-verified
//
#include <hip/hip_runtime.h>

// ---------------------------------------------------------------------------
// TwoStepPredictor for MI455X (gfx1250, wave32)
//   Stage 1: out = sigmoid(obs @ W^T + b)              (fp32 WMMA GEMM)
//   Stage 2: blocked forward substitution over DIM:
//     for block s: out_blk += out_prev @ L_blk^T       (fp32 WMMA GEMM)
//                  in-block serial substitution        (LDS + VALU)
//
// fp32 throughout via V_WMMA_F32_16X16X4_F32 (the triangular recurrence
// amplifies input rounding geometrically -> bf16/f16 WMMA unsafe).
// Tiles are staged with GLOBAL_LOAD_ASYNC_TO_LDS_B128 (ASYNCcnt) into
// double-buffered LDS so global prefetch overlaps WMMA compute.
// ---------------------------------------------------------------------------

#define DIMN 1024
#define BATCH 4096
#define KB 64          // solve block width (DIMN / KB = 16 serial stages)
#define BM 128         // GEMM workgroup tile rows (32 per wave)
#define BN 64          // GEMM workgroup tile cols
#define BKC 32         // GEMM K chunk

typedef __attribute__((ext_vector_type(2))) float v2f;
typedef __attribute__((ext_vector_type(8))) float v8f;

__device__ __forceinline__ v8f wmma_f32_16x16x4(v2f a, v2f b, v8f c) {
  // (neg_a, A, neg_b, B, c_mod, C, reuse_a, reuse_b)
  return __builtin_amdgcn_wmma_f32_16x16x4_f32(false, a, false, b, (short)0, c,
                                               false, false);
}

// Async 16-byte global -> LDS copy (per lane), tracked by ASYNCcnt.
// LDS byte address = low 32 bits of the flat pointer (ISA 10.2 aperture rule).
__device__ __forceinline__ void async_copy_b128(float* lds_dst,
                                                const float* gsrc) {
  unsigned loff = (unsigned)(unsigned long long)lds_dst;
  asm volatile("global_load_async_to_lds_b128 %0, %1, off" ::"v"(loff),
               "v"(gsrc)
               : "memory");
}
#define WAIT_ASYNC_LE_12() asm volatile("s_wait_asynccnt 12" ::: "memory")
#define WAIT_ASYNC_LE_0() asm volatile("s_wait_asynccnt 0" ::: "memory")

// Issue one BM x BKC A-tile + BN x BKC B-tile as async B128 copies.
// 128 threads: 8 B128 each for A, 4 each for B  -> 12 outstanding per thread.
__device__ __forceinline__ void issue_tile_async(
    const float* __restrict__ Ag, const float* __restrict__ Bg, int rowBase,
    int c0, int kb, float* AsBuf, float* WsBuf, int tid) {
#pragma unroll
  for (int i = 0; i < 8; ++i) {
    int idx = tid + i * 128;       // B128 index; 8 per tile row
    int row = idx >> 3;
    int col4 = (idx & 7) << 2;
    async_copy_b128(&AsBuf[row * BKC + col4],
                    &Ag[(rowBase + row) * DIMN + kb + col4]);
  }
#pragma unroll
  for (int i = 0; i < 4; ++i) {
    int idx = tid + i * 128;
    int row = idx >> 3;
    int col4 = (idx & 7) << 2;
    async_copy_b128(&WsBuf[row * BKC + col4],
                    &Bg[(c0 + row) * DIMN + kb + col4]);
  }
}

// Double-buffered WMMA mainloop. Wave tile: 32(M) x 64(N), acc[8] v8f.
// acc[t]   covers rows wave*32 + [0,16), cols c0 + t*16 + [0,16)
// acc[4+t] covers rows wave*32 + [16,32)
__device__ __forceinline__ void gemm_mainloop(const float* __restrict__ Ag,
                                              const float* __restrict__ Bg,
                                              int rowBase, int c0, int Kacc,
                                              float* As, float* Ws, v8f* acc,
                                              int tid) {
  const int wave = tid >> 5;
  const int lane = tid & 31;
  const int laneLo = lane & 15;
  const int laneHi = lane >> 4;

  const int nK = Kacc / BKC;
  issue_tile_async(Ag, Bg, rowBase, c0, 0, As, Ws, tid);

  for (int kbi = 0; kbi < nK; ++kbi) {
    float* Acur = As + (kbi & 1) * (BM * BKC);
    float* Wcur = Ws + (kbi & 1) * (BN * BKC);
    if (kbi + 1 < nK) {
      // Prefetch next chunk into the other buffer (its readers finished at
      // the trailing barrier of the previous iteration), then wait so that
      // only the 12 just-issued copies remain in flight -> current chunk done.
      issue_tile_async(Ag, Bg, rowBase, c0, (kbi + 1) * BKC,
                       As + ((kbi + 1) & 1) * (BM * BKC),
                       Ws + ((kbi + 1) & 1) * (BN * BKC), tid);
      WAIT_ASYNC_LE_12();
    } else {
      WAIT_ASYNC_LE_0();
    }
    __syncthreads();

#pragma unroll
    for (int k = 0; k < BKC; k += 4) {
      // A fragments (16x4 fp32): lane<16 -> {A[m][k],A[m][k+1]},
      //                          lane>=16 -> {A[m][k+2],A[m][k+3]}
      const float* ap0 = &Acur[(wave * 32 + laneLo) * BKC + k + 2 * laneHi];
      const float* ap1 =
          &Acur[(wave * 32 + 16 + laneLo) * BKC + k + 2 * laneHi];
      v2f a0, a1;
      a0.x = ap0[0];
      a0.y = ap0[1];
      a1.x = ap1[0];
      a1.y = ap1[1];
#pragma unroll
      for (int t = 0; t < 4; ++t) {
        // B fragment (4x16 fp32) from rows of the (transposed) weight:
        // B[kk][n] = Bg[c0 + n][kb + k + kk]
        const float* bp = &Wcur[(t * 16 + laneLo) * BKC + k + 2 * laneHi];
        v2f b;
        b.x = bp[0];
        b.y = bp[1];
        acc[t] = wmma_f32_16x16x4(a0, b, acc[t]);
        acc[4 + t] = wmma_f32_16x16x4(a1, b, acc[4 + t]);
      }
    }
    __syncthreads();
  }
}

// ---------------------------------------------------------------------------
// Kernel 1: C[r][c] = sigmoid( sum_k A[r][k] * W[c][k] + bias[c] )
// ---------------------------------------------------------------------------
__global__ __launch_bounds__(128) void gemm_bias_sigmoid_kernel(
    const float* __restrict__ A, const float* __restrict__ W,
    const float* __restrict__ bias, float* __restrict__ C) {
  __shared__ float As[2 * BM * BKC];  // 32 KB
  __shared__ float Ws[2 * BN * BKC];  // 16 KB

  const int tid = threadIdx.x;
  const int wave = tid >> 5;
  const int lane = tid & 31;
  const int laneLo = lane & 15;
  const int laneHi = lane >> 4;
  const int rowBase = blockIdx.y * BM;
  const int c0 = blockIdx.x * BN;

  v8f acc[8] = {v8f{}, v8f{}, v8f{}, v8f{}, v8f{}, v8f{}, v8f{}, v8f{}};
  gemm_mainloop(A, W, rowBase, c0, DIMN, As, Ws, acc, tid);

  // C/D layout: VGPR v, lane -> row = v + 8*laneHi, col = laneLo.
#pragma unroll
  for (int t = 0; t < 4; ++t) {
    const int col = c0 + t * 16 + laneLo;
    const float bv = bias[col];
#pragma unroll
    for (int v = 0; v < 8; ++v) {
      const int row0 = rowBase + wave * 32 + v + 8 * laneHi;
      const float x0 = acc[t][v] + bv;
      C[row0 * DIMN + col] = 1.0f / (1.0f + __expf(-x0));
      const float x1 = acc[4 + t][v] + bv;
      C[(row0 + 16) * DIMN + col] = 1.0f / (1.0f + __expf(-x1));
    }
  }
}

// ---------------------------------------------------------------------------
// Kernel 2: stage-s block GEMM update (s >= 1):
//   Out[r][s*KB + c] += sum_{k < s*KB} Out[r][k] * L[s*KB + c][k]
// ---------------------------------------------------------------------------
__global__ __launch_bounds__(128) void gemm_update_kernel(
    const float* __restrict__ Lmat, float* __restrict__ Out, int s) {
  __shared__ float As[2 * BM * BKC];
  __shared__ float Ws[2 * BN * BKC];

  const int tid = threadIdx.x;
  const int wave = tid >> 5;
  const int lane = tid & 31;
  const int laneLo = lane & 15;
  const int laneHi = lane >> 4;
  const int rowBase = blockIdx.y * BM;
  const int c0 = s * KB;  // output col base == L row base; Kacc = s*KB

  v8f acc[8] = {v8f{}, v8f{}, v8f{}, v8f{}, v8f{}, v8f{}, v8f{}, v8f{}};
  gemm_mainloop(Out, Lmat, rowBase, c0, s * KB, As, Ws, acc, tid);

#pragma unroll
  for (int t = 0; t < 4; ++t) {
    const int col = c0 + t * 16 + laneLo;
#pragma unroll
    for (int v = 0; v < 8; ++v) {
      const int row0 = rowBase + wave * 32 + v + 8 * laneHi;
      Out[row0 * DIMN + col] += acc[t][v];
      Out[(row0 + 16) * DIMN + col] += acc[4 + t][v];
    }
  }
}

// ---------------------------------------------------------------------------
// Kernel 3: in-block forward substitution for stage s.
// One thread per batch row; row-block values in padded LDS (stride 65 ->
// bank-conflict-free), KB x KB L sub-block cached in LDS. Applies the
// intervention out[:, target] = value at the correct sequential step.
// ---------------------------------------------------------------------------
__global__ __launch_bounds__(128) void block_solve_kernel(
    float* __restrict__ Out, const float* __restrict__ Lmat,
    const int* __restrict__ targetp, const float* __restrict__ valuep, int s) {
  __shared__ float Ls[KB * KB];         // 16 KB
  __shared__ float Vs[128 * (KB + 1)];  // 32.5 KB, padded stride

  const int tid = threadIdx.x;
  const int row = blockIdx.x * 128 + tid;
  const int base = s * KB;
  const int target = *targetp;
  const float val = *valuep;

  // Cache L[base..base+KB)[base..base+KB): 4096 floats, 32 per thread.
#pragma unroll
  for (int i = 0; i < 32; ++i) {
    int idx = tid + i * 128;
    int li = idx >> 6;
    int lj = idx & 63;
    Ls[idx] = Lmat[(base + li) * DIMN + base + lj];
  }
  // Stage this row's block values (currently noise + prior-block updates).
#pragma unroll
  for (int j = 0; j < KB; ++j) {
    Vs[tid * (KB + 1) + j] = Out[row * DIMN + base + j];
  }
  __syncthreads();

  float* v = &Vs[tid * (KB + 1)];
  for (int i = 0; i < KB; ++i) {
    float x;
    if (base + i == target) {
      x = val;
    } else {
      x = v[i];
      for (int j = 0; j < i; ++j) {
        x += v[j] * Ls[i * KB + j];
      }
    }
    v[i] = x;
    Out[row * DIMN + base + i] = x;
  }
}

// ---------------------------------------------------------------------------
// Launch: 1 GEMM + 16 sequential stages (GEMM update + in-block solve).
// noise is materialized directly into d_out and solved in place.
// ---------------------------------------------------------------------------
extern "C" void kernel_launch(void* const* d_in, const int* in_sizes, int n_in,
                              void* d_out, int out_size, void* d_ws,
                              size_t ws_size, hipStream_t stream) {
  (void)in_sizes; (void)n_in; (void)out_size; (void)d_ws; (void)ws_size;

  const float* obs = (const float*)d_in[0];   // [B, DIM]
  const float* W = (const float*)d_in[1];     // [DIM, DIM]
  const float* bias = (const float*)d_in[2];  // [DIM]
  const float* Lm = (const float*)d_in[3];    // [DIM, DIM] strictly lower-tri
  const float* valp = (const float*)d_in[4];  // [1] float
  const int* tgtp = (const int*)d_in[5];      // [1] int
  float* out = (float*)d_out;                 // [B, DIM]

  // Stage 1: out = sigmoid(obs @ W^T + b)
  {
    dim3 grid(DIMN / BN, BATCH / BM);
    gemm_bias_sigmoid_kernel<<<grid, 128, 0, stream>>>(obs, W, bias, out);
  }

  // Stage 2: blocked forward substitution, 16 serial stages.
  const int nblk = DIMN / KB;
  for (int s = 0; s < nblk; ++s) {
    if (s > 0) {
      dim3 grid(1, BATCH / BM);
      gemm_update_kernel<<<grid, 128, 0, stream>>>(Lm, out, s);
    }
    block_solve_kernel<<<BATCH / 128, 128, 0, stream>>>(out, Lm, tgtp, valp, s);
  }
}